// Model_42417097016597
// MI455X (gfx1250) — compile-verified
//
#include <hip/hip_runtime.h>
#include <hip/hip_bf16.h>
#include <stdint.h>

// ---------------- problem constants (match reference) ----------------
#define B_       8
#define A_       9
#define C_       80
#define H_       64
#define W_       64
#define NPB      (A_*C_*H_*W_)        // 2,949,120 floats per batch
#define TOPN     1000
#define THRESH   0.05f
#define STRIDE_F 8.0f
#define CLAMP_M  511.0f               // W*stride - 1

// ---------------- selection constants ----------------
#define BINS     (1u << 18)           // 18-bit radix histogram
#define BINSHIFT 14                   // key >> 14
#define SORTN    8192                 // candidate cap (uniform data -> ~4K worst)
#define CHUNK_F  65536                // floats per block
#define TILE_F   4096                 // floats per async tile (16 KB)
#define NTILES   (CHUNK_F / TILE_F)   // 16
#define NBLK     (NPB / CHUNK_F)      // 45 (exact)

// LDS byte offset of a __shared__ object: flat shared addresses carry the LDS
// offset in their low 32 bits (ISA 10.2 aperture table).
__device__ __forceinline__ unsigned ldsOffsetOf(const void* p) {
  return (unsigned)(size_t)p;
}

// CDNA5 async copy: global -> LDS, 16B per lane, tracked by ASYNCcnt.
// GVS form: mem_addr = SADDR(64b SGPR) + VADDR(32b byte offset) ; VDST = LDS byte addr.
__device__ __forceinline__ void asyncLoadB128(unsigned ldsByte, unsigned gByteOff,
                                              const void* sBase) {
  asm volatile("global_load_async_to_lds_b128 %0, %1, %2"
               :: "v"(ldsByte), "v"(gByteOff), "s"(sBase)
               : "memory");
}
__device__ __forceinline__ void waitAsync4() {
  asm volatile("s_wait_asynccnt 4" ::: "memory");
}
__device__ __forceinline__ void waitAsync0() {
  asm volatile("s_wait_asynccnt 0" ::: "memory");
}

// Issue one 16 KB tile: 4 async b128 per thread-slot (256 threads * 4 * 16B).
__device__ __forceinline__ void issueTile(const void* base, unsigned tileFloat0,
                                          unsigned ldsBase, int tid) {
#pragma unroll
  for (int q = 0; q < 4; ++q) {
    unsigned slot = (unsigned)(q * 256 + tid);
    asyncLoadB128(ldsBase + slot * 16u, (tileFloat0 + slot * 4u) * 4u, base);
  }
}

// ---------------- pass 1: radix histogram over async-staged tiles ----------------
__global__ __launch_bounds__(256) void k_hist(const float* __restrict__ cls,
                                              unsigned* __restrict__ hist) {
  const int b   = blockIdx.y;
  const int tid = threadIdx.x;
  const void* base = (const void*)(cls + (size_t)b * NPB);
  unsigned* h = hist + (size_t)b * BINS;

  __shared__ float4 buf[2][1024];
  const unsigned lds0 = ldsOffsetOf(&buf[0][0]);
  const unsigned lds1 = ldsOffsetOf(&buf[1][0]);
  const unsigned chunk0 = (unsigned)blockIdx.x * CHUNK_F;

  issueTile(base, chunk0, lds0, tid);           // prologue: tile 0 in flight
  for (int t = 0; t < NTILES; ++t) {
    if (t + 1 < NTILES) {
      issueTile(base, chunk0 + (unsigned)(t + 1) * TILE_F,
                ((t + 1) & 1) ? lds1 : lds0, tid);
      waitAsync4();                             // retire current tile, keep next in flight
    } else {
      waitAsync0();
    }
    const float4* bp = &buf[t & 1][0];
#pragma unroll
    for (int q = 0; q < 4; ++q) {
      float4 v = bp[q * 256 + tid];             // ds_load_b128 (own slot, no barrier needed)
      float f[4] = {v.x, v.y, v.z, v.w};
#pragma unroll
      for (int c = 0; c < 4; ++c)
        if (f[c] >= THRESH)
          atomicAdd(&h[__float_as_uint(f[c]) >> BINSHIFT], 1u);
    }
  }
}

// ---------------- pass 2: suffix-scan histogram -> cutoff prefix P1 ----------------
__global__ __launch_bounds__(1024) void k_scan(const unsigned* __restrict__ hist,
                                               unsigned* __restrict__ p1out) {
  const int b = blockIdx.x;
  const unsigned* h = hist + (size_t)b * BINS;
  __shared__ unsigned segSum[1024];
  unsigned s = 0;
  const int base = threadIdx.x * (BINS / 1024); // 256 bins per thread
  for (int i = 0; i < (int)(BINS / 1024); ++i) s += h[base + i];
  segSum[threadIdx.x] = s;
  __syncthreads();
  if (threadIdx.x == 0) {
    unsigned acc = 0, above = 0, p1 = 0;
    int seg = -1;
    for (int i = 1023; i >= 0; --i) {
      if (acc + segSum[i] >= TOPN) { seg = i; above = acc; break; }
      acc += segSum[i];
    }
    if (seg >= 0) {
      unsigned c = above;
      for (int bin = seg * 256 + 255; bin >= seg * 256; --bin) {
        c += h[bin];
        if (c >= TOPN) { p1 = (unsigned)bin; break; }
      }
    }                                           // else fewer than TOPN valid -> P1=0 (all pass)
    p1out[b] = p1;
  }
}

// ---------------- pass 3: compact candidates (key >= P1<<14) ----------------
__global__ __launch_bounds__(256) void k_compact(const float* __restrict__ cls,
                                                 const unsigned* __restrict__ p1in,
                                                 unsigned* __restrict__ candCount,
                                                 unsigned long long* __restrict__ cand) {
  const int b   = blockIdx.y;
  const int tid = threadIdx.x;
  const unsigned P1 = p1in[b];
  const void* base = (const void*)(cls + (size_t)b * NPB);
  unsigned long long* cb = cand + (size_t)b * SORTN;

  __shared__ float4 buf[2][1024];
  const unsigned lds0 = ldsOffsetOf(&buf[0][0]);
  const unsigned lds1 = ldsOffsetOf(&buf[1][0]);
  const unsigned chunk0 = (unsigned)blockIdx.x * CHUNK_F;

  issueTile(base, chunk0, lds0, tid);
  for (int t = 0; t < NTILES; ++t) {
    if (t + 1 < NTILES) {
      issueTile(base, chunk0 + (unsigned)(t + 1) * TILE_F,
                ((t + 1) & 1) ? lds1 : lds0, tid);
      waitAsync4();
    } else {
      waitAsync0();
    }
    const float4* bp = &buf[t & 1][0];
#pragma unroll
    for (int q = 0; q < 4; ++q) {
      float4 v = bp[q * 256 + tid];
      float f[4] = {v.x, v.y, v.z, v.w};
      unsigned idx0 = chunk0 + (unsigned)t * TILE_F + (unsigned)(q * 256 + tid) * 4u;
#pragma unroll
      for (int c = 0; c < 4; ++c) {
        if (f[c] >= THRESH) {
          unsigned key = __float_as_uint(f[c]);
          if ((key >> BINSHIFT) >= P1) {
            unsigned pos = atomicAdd(&candCount[b], 1u);
            if (pos < SORTN)
              cb[pos] = ((unsigned long long)key << 32) | (unsigned)(~(idx0 + c));
          }
        }
      }
    }
  }
}

// ---------------- pass 4: per-batch bitonic sort (LDS) + box decode ----------------
__global__ __launch_bounds__(1024) void k_sort_decode(
    const unsigned long long* __restrict__ cand,
    const unsigned* __restrict__ candCount,
    const float* __restrict__ box,
    const float* __restrict__ anchors,
    float* __restrict__ out) {
  const int b = blockIdx.x;
  __shared__ unsigned long long s[SORTN];       // 64 KB of the 320 KB WGP LDS
  unsigned count = candCount[b];
  if (count > SORTN) count = SORTN;
  const unsigned long long* cb = cand + (size_t)b * SORTN;
  for (int e = threadIdx.x; e < SORTN; e += 1024)
    s[e] = (e < (int)count) ? cb[e] : 0ULL;     // 0 sorts to the end (keys are large)
  __syncthreads();

  for (int k = 2; k <= SORTN; k <<= 1) {
    for (int j = k >> 1; j > 0; j >>= 1) {
      for (int e = threadIdx.x; e < SORTN; e += 1024) {
        int ix = e ^ j;
        if (ix > e) {
          unsigned long long a = s[e], bb = s[ix];
          bool desc = ((e & k) == 0);
          if ((a < bb) == desc) { s[e] = bb; s[ix] = a; }
        }
      }
      __syncthreads();
    }
  }

  if (threadIdx.x < TOPN) {
    const int jo = threadIdx.x;
    const unsigned long long v = s[jo];
    float score = 0.0f, clsOut = 0.0f, o0 = 0.0f, o1 = 0.0f, o2 = 0.0f, o3 = 0.0f;
    if (v != 0ULL) {
      const unsigned key = (unsigned)(v >> 32);
      const unsigned idx = ~(unsigned)(v & 0xFFFFFFFFull);
      score = __uint_as_float(key);
      const int x = (int)(idx & 63u);
      const int y = (int)((idx >> 6) & 63u);
      const int c = (int)((idx >> 12) % (unsigned)C_);
      const int a = (int)(idx / (unsigned)(C_ * H_ * W_));
      const size_t bo = ((size_t)b * (A_ * 4) + (size_t)a * 4) * (H_ * W_) + (y << 6) + x;
      const float d0 = box[bo];
      const float d1 = box[bo + 1 * H_ * W_];
      const float d2 = box[bo + 2 * H_ * W_];
      const float d3 = box[bo + 3 * H_ * W_];
      const float gx = (float)x * STRIDE_F, gy = (float)y * STRIDE_F;
      const float g0 = gx + anchors[a * 4 + 0];
      const float g1 = gy + anchors[a * 4 + 1];
      const float g2 = gx + anchors[a * 4 + 2];
      const float g3 = gy + anchors[a * 4 + 3];
      const float aw = g2 - g0 + 1.0f, ah = g3 - g1 + 1.0f;
      const float cx = g0 + 0.5f * aw, cy = g1 + 0.5f * ah;
      const float px = d0 * aw + cx, py = d1 * ah + cy;
      const float pw = expf(d2) * aw, ph = expf(d3) * ah;
      o0 = fmaxf(0.0f, fminf(px - 0.5f * pw, CLAMP_M));
      o1 = fmaxf(0.0f, fminf(py - 0.5f * ph, CLAMP_M));
      o2 = fmaxf(0.0f, fminf(px + 0.5f * pw - 1.0f, CLAMP_M));
      o3 = fmaxf(0.0f, fminf(py + 0.5f * ph - 1.0f, CLAMP_M));
      clsOut = (float)c;
    }
    // output: scores [0,8000) | boxes [8000,40000) | classes [40000,48000)
    out[b * TOPN + jo] = score;
    float* ob = out + B_ * TOPN + (size_t)(b * TOPN + jo) * 4;
    ob[0] = o0; ob[1] = o1; ob[2] = o2; ob[3] = o3;
    out[B_ * TOPN * 5 + b * TOPN + jo] = clsOut;
  }
}

// ---------------- host entry ----------------
extern "C" void kernel_launch(void* const* d_in, const int* in_sizes, int n_in,
                              void* d_out, int out_size, void* d_ws, size_t ws_size,
                              hipStream_t stream) {
  (void)in_sizes; (void)n_in; (void)out_size; (void)ws_size;
  const float* cls  = (const float*)d_in[0];
  const float* box  = (const float*)d_in[1];
  const float* anch = (const float*)d_in[2];
  float* out = (float*)d_out;

  // ws layout: hist[8][BINS] u32 | candCount[8] u32 | P1[8] u32 | pad | cand[8][SORTN] u64
  const size_t histBytes = (size_t)B_ * BINS * sizeof(unsigned);   // 8 MB
  unsigned* hist       = (unsigned*)d_ws;
  unsigned* candCount  = (unsigned*)((char*)d_ws + histBytes);
  unsigned* p1         = candCount + B_;
  unsigned long long* cand = (unsigned long long*)((char*)d_ws + histBytes + 64);

  // re-zero scratch every call (graph-capture safe, deterministic replays)
  hipMemsetAsync(d_ws, 0, histBytes + 64, stream);

  k_hist       <<<dim3(NBLK, B_), 256, 0, stream>>>(cls, hist);
  k_scan       <<<B_, 1024, 0, stream>>>(hist, p1);
  k_compact    <<<dim3(NBLK, B_), 256, 0, stream>>>(cls, p1, candCount, cand);
  k_sort_decode<<<B_, 1024, 0, stream>>>(cand, candCount, box, anch, out);
}